// GCNLayer_89988154785839
// MI455X (gfx1250) — compile-verified
//
#include <hip/hip_runtime.h>
#include <hip/hip_bf16.h>

typedef __attribute__((ext_vector_type(2))) float v2f;
typedef __attribute__((ext_vector_type(8))) float v8f;

#define D 128           // D_IN == D_OUT == 128
#define NNODES 40000

// ---------------------------------------------------------------------------
// Kernel 1: zero the accumulation workspace (summed[N*D] ++ cnt[N]).
// ---------------------------------------------------------------------------
__global__ void gcn_zero_kernel(float* __restrict__ p, int n) {
    int i = blockIdx.x * blockDim.x + threadIdx.x;
    if (i < n) p[i] = 0.0f;
}

// ---------------------------------------------------------------------------
// Kernel 2: edge phase. One wave32 per edge; lane l owns float4 chunk l of the
// 128-wide feature row. msg = feat[src]*feat[dst], atomically accumulated into
// summed[dst]; lane 0 counts the edge. Feature table (20.5MB) + summed (20.5MB)
// are L2-resident on MI455X (192MB L2), so gathers/atomics stay on-chip.
// ---------------------------------------------------------------------------
__global__ void gcn_edge_kernel(const float* __restrict__ feat,
                                const int* __restrict__ src,
                                const int* __restrict__ dst,
                                float* __restrict__ summed,
                                float* __restrict__ cnt,
                                int n_edges) {
    int gid  = blockIdx.x * blockDim.x + threadIdx.x;
    int edge = gid >> 5;          // whole wave shares one edge
    int lane = gid & 31;
    if (edge >= n_edges) return;

    int s = src[edge];
    int d = dst[edge];

    const float4 a = ((const float4*)(feat + (size_t)s * D))[lane];
    const float4 b = ((const float4*)(feat + (size_t)d * D))[lane];

    float* row = summed + (size_t)d * D + lane * 4;
    atomicAdd(row + 0, a.x * b.x);
    atomicAdd(row + 1, a.y * b.y);
    atomicAdd(row + 2, a.z * b.z);
    atomicAdd(row + 3, a.w * b.w);
    if (lane == 0) atomicAdd(cnt + d, 1.0f);
}

// ---------------------------------------------------------------------------
// Kernel 3: mean normalization in place: x[n][d] = summed[n][d]/max(cnt[n],1)
// ---------------------------------------------------------------------------
__global__ void gcn_norm_kernel(float* __restrict__ summed,
                                const float* __restrict__ cnt,
                                int total) {
    int i = blockIdx.x * blockDim.x + threadIdx.x;
    if (i < total) {
        int n = i >> 7;   // /128
        summed[i] = summed[i] / fmaxf(cnt[n], 1.0f);
    }
}

// ---------------------------------------------------------------------------
// Kernel 4: out = x @ W^T + b via V_WMMA_F32_16X16X4_F32 (fp32 matrix path —
// matches reference precision; GEMM is bandwidth-trivial at 1.3 GFLOP).
// Block = 256 threads = 8 waves. Block owns a 16-row node strip; wave w owns
// output columns [16w, 16w+16). 32 WMMA steps cover K=128.
//
// Fragment layouts per CDNA5 ISA (32-bit operands, wave32):
//   A 16x4:  lanes 0-15: M=lane,  VGPR0=K0,  VGPR1=K1 ; lanes 16-31: K2,K3
//   B 4x16:  lanes 0-15: N=lane,  VGPR0=K0,  VGPR1=K1 ; lanes 16-31: K2,K3
//   C/D 16x16: VGPR j -> row j (lanes 0-15) / row j+8 (lanes 16-31), N=lane%16
// ---------------------------------------------------------------------------
__global__ __launch_bounds__(256) void gcn_gemm_kernel(
        const float* __restrict__ x,     // [N, 128] normalized features
        const float* __restrict__ W,     // [128, 128] row-major (D_OUT, D_IN)
        const float* __restrict__ bias,  // [128]
        float* __restrict__ out) {       // [N, 128]
    const int wave  = threadIdx.x >> 5;     // 0..7 -> column tile
    const int lane  = threadIdx.x & 31;
    const int lhalf = lane & 15;
    const int khi   = (lane >> 4) << 1;     // 0 for lanes 0-15, 2 for 16-31
    const int m0    = blockIdx.x * 16;      // node strip base
    const int o0    = wave * 16;            // output-column tile base

    // A: row (m0+lhalf) of x ; B: B[k][o] = W[o*128+k] -> row (o0+lhalf) of W
    const float* arow = x + (size_t)(m0 + lhalf) * D + khi;
    const float* brow = W + (size_t)(o0 + lhalf) * D + khi;

    v8f c = {};
#pragma unroll
    for (int kk = 0; kk < 32; ++kk) {
        v2f a = *(const v2f*)(arow + kk * 4);
        v2f b = *(const v2f*)(brow + kk * 4);
        c = __builtin_amdgcn_wmma_f32_16x16x4_f32(
                /*neg_a=*/false, a, /*neg_b=*/false, b,
                /*c_mod=*/(short)0, c, /*reuse_a=*/false, /*reuse_b=*/false);
    }

    const float bv    = bias[o0 + lhalf];
    const int   mbase = m0 + ((lane >> 4) << 3);   // +8 rows for upper half
#pragma unroll
    for (int j = 0; j < 8; ++j) {
        out[(size_t)(mbase + j) * D + o0 + lhalf] = c[j] + bv;
    }
}

// ---------------------------------------------------------------------------
extern "C" void kernel_launch(void* const* d_in, const int* in_sizes, int n_in,
                              void* d_out, int out_size, void* d_ws, size_t ws_size,
                              hipStream_t stream) {
    const float* feat = (const float*)d_in[0];   // [N, 128]
    const int*   src  = (const int*)  d_in[1];   // [E]
    const int*   dst  = (const int*)  d_in[2];   // [E]
    const float* W    = (const float*)d_in[3];   // [128, 128]
    const float* bias = (const float*)d_in[4];   // [128]
    float*       out  = (float*)d_out;

    const int n_nodes = in_sizes[0] / D;         // 40000
    const int n_edges = in_sizes[1];             // 640000

    float* summed = (float*)d_ws;                        // [N * 128]
    float* cnt    = summed + (size_t)n_nodes * D;        // [N]

    // 1) zero summed + cnt
    {
        int total = n_nodes * D + n_nodes;
        gcn_zero_kernel<<<(total + 255) / 256, 256, 0, stream>>>(summed, total);
    }
    // 2) edge scatter-accumulate (one wave per edge)
    {
        long long threads = (long long)n_edges * 32;
        int blocks = (int)((threads + 255) / 256);
        gcn_edge_kernel<<<blocks, 256, 0, stream>>>(feat, src, dst, summed, cnt,
                                                    n_edges);
    }
    // 3) mean normalize in place
    {
        int total = n_nodes * D;
        gcn_norm_kernel<<<(total + 255) / 256, 256, 0, stream>>>(summed, cnt, total);
    }
    // 4) WMMA GEMM: out = x @ W^T + b   (N divisible by 16 -> no tail)
    {
        int blocks = n_nodes / 16;   // 2500; 8 waves/block cover all 128 cols
        gcn_gemm_kernel<<<blocks, 256, 0, stream>>>(summed, W, bias, out);
    }
}